// GNNPredictorWithDistance_74217034875600
// MI455X (gfx1250) — compile-verified
//
#include <hip/hip_runtime.h>
#include <hip/hip_bf16.h>

typedef __attribute__((ext_vector_type(2))) float v2f;
typedef __attribute__((ext_vector_type(8))) float v8f;

#define N_NODES 16384     // B*T
#define HID     128
#define LFEAT   64
#define NEDGE   524288
#define TSTEPS  256
#define BATCH   64

// LDS byte-offset of a shared-memory pointer (low 32 bits of the flat address
// are the LDS address per the CDNA5 aperture mapping).
__device__ __forceinline__ unsigned lds_addr(const void* p) {
    return (unsigned)(uintptr_t)p;
}
// 16B async copy global -> LDS (per-lane), tracked by ASYNCcnt.
__device__ __forceinline__ void async_copy16(unsigned lds_off, const void* gptr) {
    asm volatile("global_load_async_to_lds_b128 %0, %1, off"
                 :: "v"(lds_off), "v"(gptr) : "memory");
}
__device__ __forceinline__ void wait_async0() {
    asm volatile("s_wait_asynccnt 0x0" ::: "memory");
}

// ---------------------------------------------------------------------------
// small helpers
// ---------------------------------------------------------------------------
__global__ void fill_kernel(float* __restrict__ p, float v, int n) {
    int i = blockIdx.x * blockDim.x + threadIdx.x;
    if (i < n) p[i] = v;
}

__global__ void deg_count_kernel(const int* __restrict__ dst, float* __restrict__ deg, int e_cnt) {
    int e = blockIdx.x * blockDim.x + threadIdx.x;
    if (e < e_cnt) atomicAdd(&deg[dst[e]], 1.0f);
}

// cdist[j] = sum_k dist[k] * W0[(64+k)*128 + j]   (broadcast distance block of feat@W0)
__global__ void dist_proj_kernel(const float* __restrict__ dist, const float* __restrict__ W0,
                                 float* __restrict__ cdist) {
    __shared__ float red[256];
    const int j = blockIdx.x;           // 0..127
    float acc = 0.f;
    for (int k = threadIdx.x; k < LFEAT * LFEAT; k += 256)
        acc += dist[k] * W0[(LFEAT + k) * HID + j];
    red[threadIdx.x] = acc;
    __syncthreads();
    for (int s = 128; s > 0; s >>= 1) {
        if (threadIdx.x < s) red[threadIdx.x] += red[threadIdx.x + s];
        __syncthreads();
    }
    if (threadIdx.x == 0) cdist[j] = red[0];
}

// ---------------------------------------------------------------------------
// fp32 WMMA GEMM: Out[M,128] = A[M,K] @ B[K,128] (+ cvec per column)
// A rows contiguous (lda == K). grid.x = M/16, 256 threads = 8 waves.
// B panel (K*128) and the block's A tile (16*K) are staged into LDS via the
// async-to-LDS engine once per block; WMMA fragments then come from LDS.
// Dynamic LDS = (K*128 + 16*K)*4 bytes.
// ---------------------------------------------------------------------------
__global__ __launch_bounds__(256)
void gemm_wmma_f32(const float* __restrict__ A,
                   const float* __restrict__ B,
                   const float* __restrict__ cvec,
                   float* __restrict__ Out, int K) {
    extern __shared__ float lds[];
    float* sB = lds;               // [K,128]
    float* sA = lds + K * HID;     // [16,K]

    const int tid = threadIdx.x;
    const int m0 = blockIdx.x * 16;

    {   // async stage: weights + A tile
        const char* gB = (const char*)B;
        const char* gA = (const char*)(A + (size_t)m0 * K);
        const unsigned lB = lds_addr(sB);
        const unsigned lA = lds_addr(sA);
        const int bBytes = K * HID * 4;     // 32KB or 64KB (multiple of 4096)
        const int aBytes = 16 * K * 4;      // 4KB or 8KB
        for (int ofs = tid * 16; ofs < bBytes; ofs += 256 * 16)
            async_copy16(lB + ofs, gB + ofs);
        for (int ofs = tid * 16; ofs < aBytes; ofs += 256 * 16)
            async_copy16(lA + ofs, gA + ofs);
        wait_async0();
    }
    __syncthreads();

    const int lane = tid & 31;
    const int wave = tid >> 5;
    const int n0 = wave * 16;
    const int mn = lane & 15;            // m for A-frag, n for B-frag / C column
    const int kb = (lane >> 4) << 1;     // 0 or 2

    v8f acc = {};
    for (int kk = 0; kk < K; kk += 4) {
        v2f a, b;
        a.x = sA[mn * K + kk + kb];
        a.y = sA[mn * K + kk + kb + 1];
        b.x = sB[(kk + kb) * HID + n0 + mn];
        b.y = sB[(kk + kb + 1) * HID + n0 + mn];
        acc = __builtin_amdgcn_wmma_f32_16x16x4_f32(false, a, false, b,
                                                    (short)0, acc, false, false);
    }
    const float cadd = cvec ? cvec[n0 + mn] : 0.0f;
    const int mbase = m0 + ((lane >> 4) << 3);
#pragma unroll
    for (int r = 0; r < 8; ++r)
        Out[(mbase + r) * HID + n0 + mn] = acc[r] + cadd;
}

// ---------------------------------------------------------------------------
// Edge aggregation: agg[dst] += h[src] * rsqrt(deg[src])*rsqrt(deg[dst])
// 32 lanes per edge, 4 floats per lane (coalesced 512B row read, f32 atomics in L2)
// ---------------------------------------------------------------------------
__global__ __launch_bounds__(256)
void edge_agg_kernel(const int* __restrict__ src, const int* __restrict__ dst,
                     const float* __restrict__ deg, const float* __restrict__ h,
                     float* __restrict__ agg, int e_cnt) {
    long long tid = (long long)blockIdx.x * blockDim.x + threadIdx.x;
    long long e = tid >> 5;
    if (e >= e_cnt) return;
    const int lane = threadIdx.x & 31;
    const int s = src[e];
    const int d = dst[e];
    const float norm = rsqrtf(deg[s]) * rsqrtf(deg[d]);
    const float4 hv = *(const float4*)(h + (long long)s * HID + lane * 4);
    float* ap = agg + (long long)d * HID + lane * 4;
    atomicAdd(ap + 0, hv.x * norm);
    atomicAdd(ap + 1, hv.y * norm);
    atomicAdd(ap + 2, hv.z * norm);
    atomicAdd(ap + 3, hv.w * norm);
}

// out = relu(agg + h * (1/deg) + bias[col])
__global__ void self_relu_kernel(const float* __restrict__ agg, const float* __restrict__ h,
                                 const float* __restrict__ deg, const float* __restrict__ bias,
                                 float* __restrict__ out, int n_elems) {
    int i = blockIdx.x * blockDim.x + threadIdx.x;
    if (i >= n_elems) return;
    const int n = i >> 7;
    const int j = i & (HID - 1);
    const float dinv2 = 1.0f / deg[n];   // (deg^-1/2)^2
    const float v = agg[i] + h[i] * dinv2 + bias[j];
    out[i] = v > 0.f ? v : 0.f;
}

// ---------------------------------------------------------------------------
// Persistent-block LSTM layer. 1024 threads = 32 waves. LDS:
//   h_s[64*128] c_s[64*128] g_s[64*512] x_s[2][64*128]  -> 256 KB.
// x_t is staged into LDS with the async engine, double-buffered: the copy of
// x_{t+1} is issued before the WMMA work of step t and drained at step end
// (s_wait_asynccnt 0 + barrier), overlapping DMA with matrix math.
//   x_t(b,k) = X[t*tStride + b*bStride + k]
// ---------------------------------------------------------------------------
__global__ __launch_bounds__(1024)
void lstm_layer_kernel(const float* __restrict__ X, int tStride, int bStride,
                       const float* __restrict__ Wih, const float* __restrict__ Whh,
                       const float* __restrict__ bih, const float* __restrict__ bhh,
                       float* __restrict__ ys, float* __restrict__ lastH) {
    extern __shared__ float smem[];
    float* h_s = smem;                        // 64*128
    float* c_s = h_s + BATCH * HID;           // 64*128
    float* g_s = c_s + BATCH * HID;           // 64*512
    float* x_s[2] = { g_s + BATCH * 4 * HID,                // 64*128
                      g_s + BATCH * 4 * HID + BATCH * HID };// 64*128

    const int tid = threadIdx.x;
    for (int i = tid; i < BATCH * HID; i += blockDim.x) { h_s[i] = 0.f; c_s[i] = 0.f; }

    // preload x_0 (each 16B chunk: row = batch, col = 4-float group)
    {
        const unsigned lx = lds_addr(x_s[0]);
        for (int c = tid; c < BATCH * 32; c += 1024) {
            const int row = c >> 5;
            const int col = (c & 31) << 2;
            async_copy16(lx + ((row << 7) + col) * 4,
                         X + (size_t)row * bStride + col);
        }
        wait_async0();
    }
    __syncthreads();

    const int lane = tid & 31;
    const int wave = tid >> 5;               // 0..31
    const int mn = lane & 15;
    const int kb = (lane >> 4) << 1;

    int cb = 0;
    for (int t = 0; t < TSTEPS; ++t) {
        // issue async copy of x_{t+1} into the other buffer
        if (t + 1 < TSTEPS) {
            const unsigned lx = lds_addr(x_s[cb ^ 1]);
            const float* Xn = X + (size_t)(t + 1) * tStride;
            for (int c = tid; c < BATCH * 32; c += 1024) {
                const int row = c >> 5;
                const int col = (c & 31) << 2;
                async_copy16(lx + ((row << 7) + col) * 4,
                             Xn + (size_t)row * bStride + col);
            }
        }
        const float* xt = x_s[cb];
        // 4 M-tiles x 32 N-tiles of the [64,512] gate matrix; each wave takes 4 tiles
        for (int q = 0; q < 4; ++q) {
            const int tile = wave * 4 + q;   // 0..127
            const int m0 = (tile & 3) * 16;
            const int n0 = (tile >> 2) * 16;
            v8f acc = {};
            // x_t @ Wih^T   (B[k][n] = Wih[n*128 + k]; A from LDS)
            for (int kk = 0; kk < HID; kk += 4) {
                v2f a, b;
                a.x = xt[(m0 + mn) * HID + kk + kb];
                a.y = xt[(m0 + mn) * HID + kk + kb + 1];
                b.x = Wih[(n0 + mn) * HID + kk + kb];
                b.y = Wih[(n0 + mn) * HID + kk + kb + 1];
                acc = __builtin_amdgcn_wmma_f32_16x16x4_f32(false, a, false, b,
                                                            (short)0, acc, false, false);
            }
            // h @ Whh^T  (h from LDS)
            for (int kk = 0; kk < HID; kk += 4) {
                v2f a, b;
                a.x = h_s[(m0 + mn) * HID + kk + kb];
                a.y = h_s[(m0 + mn) * HID + kk + kb + 1];
                b.x = Whh[(n0 + mn) * HID + kk + kb];
                b.y = Whh[(n0 + mn) * HID + kk + kb + 1];
                acc = __builtin_amdgcn_wmma_f32_16x16x4_f32(false, a, false, b,
                                                            (short)0, acc, false, false);
            }
            const int mbase = m0 + ((lane >> 4) << 3);
#pragma unroll
            for (int r = 0; r < 8; ++r)
                g_s[(mbase + r) * (4 * HID) + n0 + mn] = acc[r];
        }
        __syncthreads();
        // fused gates: i,f,g,o blocks of 128 columns each
        for (int i = tid; i < BATCH * HID; i += blockDim.x) {
            const int b = i >> 7;
            const int j = i & (HID - 1);
            const float* gr = g_s + b * (4 * HID);
            float gi = gr[j]           + bih[j]           + bhh[j];
            float gf = gr[HID + j]     + bih[HID + j]     + bhh[HID + j];
            float gg = gr[2 * HID + j] + bih[2 * HID + j] + bhh[2 * HID + j];
            float go = gr[3 * HID + j] + bih[3 * HID + j] + bhh[3 * HID + j];
            gi = 1.0f / (1.0f + __expf(-gi));
            gf = 1.0f / (1.0f + __expf(-gf));
            go = 1.0f / (1.0f + __expf(-go));
            gg = tanhf(gg);
            const float c = gf * c_s[i] + gi * gg;
            const float h = go * tanhf(c);
            c_s[i] = c;
            h_s[i] = h;
            if (ys) ys[(long long)t * BATCH * HID + i] = h;
            if (lastH && t == TSTEPS - 1) lastH[i] = h;
        }
        wait_async0();       // x_{t+1} resident
        __syncthreads();
        cb ^= 1;
    }
}

// ---------------------------------------------------------------------------
// FC: Out[64,768] = A[64,128] @ W[768,128]^T + bias. 192 tiles, 24 blocks x 8 waves.
// ---------------------------------------------------------------------------
__global__ __launch_bounds__(256)
void fc_wmma_kernel(const float* __restrict__ A, const float* __restrict__ W,
                    const float* __restrict__ bias, float* __restrict__ Out) {
    const int lane = threadIdx.x & 31;
    const int wave = threadIdx.x >> 5;
    const int tile = blockIdx.x * 8 + wave;  // 0..191
    const int m0 = (tile & 3) * 16;
    const int n0 = (tile >> 2) * 16;
    const int mn = lane & 15;
    const int kb = (lane >> 4) << 1;
    v8f acc = {};
    for (int kk = 0; kk < HID; kk += 4) {
        v2f a, b;
        a.x = A[(m0 + mn) * HID + kk + kb];
        a.y = A[(m0 + mn) * HID + kk + kb + 1];
        b.x = W[(n0 + mn) * HID + kk + kb];   // B[k][n] = fc_W[n][k]
        b.y = W[(n0 + mn) * HID + kk + kb + 1];
        acc = __builtin_amdgcn_wmma_f32_16x16x4_f32(false, a, false, b,
                                                    (short)0, acc, false, false);
    }
    const float bb = bias[n0 + mn];
    const int mbase = m0 + ((lane >> 4) << 3);
#pragma unroll
    for (int r = 0; r < 8; ++r)
        Out[(mbase + r) * 768 + n0 + mn] = acc[r] + bb;
}

// ---------------------------------------------------------------------------
extern "C" void kernel_launch(void* const* d_in, const int* in_sizes, int n_in,
                              void* d_out, int out_size, void* d_ws, size_t ws_size,
                              hipStream_t stream) {
    const float* x        = (const float*)d_in[0];   // [64,256,64] = [N,64]
    const float* distance = (const float*)d_in[1];   // [64,64]
    const int*   ei       = (const int*)  d_in[2];   // [2,E]
    const float* W0  = (const float*)d_in[3];
    const float* b0  = (const float*)d_in[4];
    const float* W1  = (const float*)d_in[5];
    const float* b1  = (const float*)d_in[6];
    const float* W2  = (const float*)d_in[7];
    const float* b2  = (const float*)d_in[8];
    const float* Wih0 = (const float*)d_in[9];
    const float* Whh0 = (const float*)d_in[10];
    const float* bih0 = (const float*)d_in[11];
    const float* bhh0 = (const float*)d_in[12];
    const float* Wih1 = (const float*)d_in[13];
    const float* Whh1 = (const float*)d_in[14];
    const float* bih1 = (const float*)d_in[15];
    const float* bhh1 = (const float*)d_in[16];
    const float* fcW  = (const float*)d_in[17];      // [768,128]
    const float* fcb  = (const float*)d_in[18];
    float* out = (float*)d_out;                      // [64,12,64] == [64,768]

    const int* srcp = ei;
    const int* dstp = ei + NEDGE;

    // workspace layout
    float* buf0  = (float*)d_ws;                     // [N,128]
    float* buf1  = buf0 + (size_t)N_NODES * HID;     // [N,128]
    float* buf2  = buf1 + (size_t)N_NODES * HID;     // [N,128] (agg)
    float* deg   = buf2 + (size_t)N_NODES * HID;     // [N]
    float* cdist = deg + N_NODES;                    // [128]
    float* ys1   = cdist + HID;                      // [T,B,128]
    float* lasth = ys1 + (size_t)TSTEPS * BATCH * HID; // [64,128]

    const int NH = N_NODES * HID;

    // degrees (self loop = 1 + in-degree)
    fill_kernel<<<(N_NODES + 255) / 256, 256, 0, stream>>>(deg, 1.0f, N_NODES);
    deg_count_kernel<<<(NEDGE + 255) / 256, 256, 0, stream>>>(dstp, deg, NEDGE);

    // constant distance-block contribution of feat @ W0
    dist_proj_kernel<<<HID, 256, 0, stream>>>(distance, W0, cdist);

    const size_t smem64  = (size_t)(LFEAT * HID + 16 * LFEAT) * sizeof(float); // 36 KB
    const size_t smem128 = (size_t)(HID * HID + 16 * HID) * sizeof(float);     // 72 KB

    // ---- GCN layer 0: h = x @ W0[:64] + cdist ----
    gemm_wmma_f32<<<N_NODES / 16, 256, smem64, stream>>>(x, W0, cdist, buf1, LFEAT);
    fill_kernel<<<(NH + 255) / 256, 256, 0, stream>>>(buf2, 0.0f, NH);
    edge_agg_kernel<<<(unsigned)((long long)NEDGE * 32 / 256), 256, 0, stream>>>(
        srcp, dstp, deg, buf1, buf2, NEDGE);
    self_relu_kernel<<<(NH + 255) / 256, 256, 0, stream>>>(buf2, buf1, deg, b0, buf0, NH);

    // ---- GCN layer 1 ----
    gemm_wmma_f32<<<N_NODES / 16, 256, smem128, stream>>>(buf0, W1, nullptr, buf1, HID);
    fill_kernel<<<(NH + 255) / 256, 256, 0, stream>>>(buf2, 0.0f, NH);
    edge_agg_kernel<<<(unsigned)((long long)NEDGE * 32 / 256), 256, 0, stream>>>(
        srcp, dstp, deg, buf1, buf2, NEDGE);
    self_relu_kernel<<<(NH + 255) / 256, 256, 0, stream>>>(buf2, buf1, deg, b1, buf0, NH);

    // ---- GCN layer 2 ----
    gemm_wmma_f32<<<N_NODES / 16, 256, smem128, stream>>>(buf0, W2, nullptr, buf1, HID);
    fill_kernel<<<(NH + 255) / 256, 256, 0, stream>>>(buf2, 0.0f, NH);
    edge_agg_kernel<<<(unsigned)((long long)NEDGE * 32 / 256), 256, 0, stream>>>(
        srcp, dstp, deg, buf1, buf2, NEDGE);
    self_relu_kernel<<<(NH + 255) / 256, 256, 0, stream>>>(buf2, buf1, deg, b2, buf0, NH);

    // ---- LSTM layers (persistent block, 256 KB dynamic LDS, async x staging) ----
    const size_t lstm_smem =
        (size_t)(2 * BATCH * HID + BATCH * 4 * HID + 2 * BATCH * HID) * sizeof(float);
    // layer 0: x_t(b,k) = buf0[(b*T + t)*128 + k]  -> tStride=128, bStride=T*128
    lstm_layer_kernel<<<1, 1024, lstm_smem, stream>>>(
        buf0, HID, TSTEPS * HID, Wih0, Whh0, bih0, bhh0, ys1, nullptr);
    // layer 1: x_t(b,k) = ys1[t*B*128 + b*128 + k] -> tStride=B*128, bStride=128
    lstm_layer_kernel<<<1, 1024, lstm_smem, stream>>>(
        ys1, BATCH * HID, HID, Wih1, Whh1, bih1, bhh1, nullptr, lasth);

    // ---- FC ----
    fc_wmma_kernel<<<24, 256, 0, stream>>>(lasth, fcW, fcb, out);
}